// GCN_15762529976410
// MI455X (gfx1250) — compile-verified
//
#include <hip/hip_runtime.h>

typedef __attribute__((ext_vector_type(2))) float v2f;
typedef __attribute__((ext_vector_type(4))) float v4f;
typedef __attribute__((ext_vector_type(8))) float v8f;

#define NROWS 4096
#define NF    128
#define KB    64
#define LDT   68   // padded dword stride of transposed LDS B-tile (8B-aligned; conflict-free b64 reads)
#define TILE  (NF * LDT)

// ---------------------------------------------------------------------------
// Kernel 1: d_inv[b*N + n] = 1/sqrt(sum_m adj[b,n,m] + 1e-6)
// One wave per row; float4 sweep -> 512B per wave per instruction.
// ---------------------------------------------------------------------------
__global__ void gcn_degree(const float* __restrict__ adj, float* __restrict__ dinv)
{
    const int  lane = threadIdx.x & 31;
    const int  wave = threadIdx.x >> 5;
    const long row  = (long)blockIdx.x * 8 + wave;        // global row in [0, B*N)
    const v4f* rp4  = (const v4f*)(adj + row * (long)NROWS);

    float s = 0.f;
    for (int k = lane; k < NROWS / 4; k += 32) {
        v4f v = rp4[k];
        s += v.x + v.y + v.z + v.w;
    }
    #pragma unroll
    for (int o = 16; o > 0; o >>= 1) s += __shfl_xor(s, o, 32);

    if (lane == 0) dinv[row] = 1.0f / sqrtf(s + 1e-6f);
}

// ---------------------------------------------------------------------------
// Fused fp32 WMMA GEMM:  out[gRow, c] = dinv[gRow] * (A_b @ B_b)[row, c] (+ bias[c])
//   A: [B][4096][lda] row-major (per-batch stride aBatchStride elements), streamed NT
//   B: [B][K][128]    row-major (per-batch stride bBatchStride; 0 = shared), L2-hot
// Grid: B * (4096/128) workgroups of 256 threads (8 waves).
// Wave w -> rows [rowBlk*128 + w*16, +16) x all 128 cols (8 v8f accumulators).
// Double-buffered transposed LDS B-panel; one barrier per K-stage.
// ---------------------------------------------------------------------------
__global__ __launch_bounds__(256, 1)
void gcn_gemm(const float* __restrict__ A, long lda, long aBatchStride,
              const float* __restrict__ Bm, long bBatchStride, int K,
              const float* __restrict__ dinv, const float* __restrict__ bias,
              float* __restrict__ out)
{
    __shared__ float bt[2 * TILE];            // bt[buf][c*LDT + r] = B[k0+r][c]

    const int tid    = threadIdx.x;
    const int lane   = tid & 31;
    const int wave   = tid >> 5;
    const int batch  = blockIdx.x >> 5;       // 32 row-blocks per batch
    const int rowBlk = blockIdx.x & 31;
    const int m0     = rowBlk * 128 + wave * 16;   // row within batch
    const int lh     = lane >> 4;             // lane half: selects K pair
    const int ll     = lane & 15;             // row (A) / col (B,C) within tile
    const long gRow0 = (long)batch * NROWS + m0;

    const float* Ap   = A  + (long)batch * aBatchStride;
    const float* Bp   = Bm + (long)batch * bBatchStride;
    const float* aRow = Ap + (long)(m0 + ll) * lda + lh * 2;  // per-lane A base

    v8f acc[8] = {};

    const int nStages = K / KB;

    // Preload stage 0 B-panel (transposed).
    #pragma unroll
    for (int i = 0; i < (KB * NF) / 256; ++i) {
        int idx = tid + i * 256;
        int r   = idx >> 7;
        int c   = idx & 127;
        bt[c * LDT + r] = Bp[(long)r * NF + c];
    }

    for (int s = 0; s < nStages; ++s) {
        __syncthreads();                       // stage-s panel ready
        const float* cur = bt + (s & 1) * TILE;

        // Prefetch stage s+1 panel into the other buffer (overlaps WMMA work).
        if (s + 1 < nStages) {
            float* nxt = bt + ((s + 1) & 1) * TILE;
            const long k0n = (long)(s + 1) * KB;
            #pragma unroll
            for (int i = 0; i < (KB * NF) / 256; ++i) {
                int idx = tid + i * 256;
                int r   = idx >> 7;
                int c   = idx & 127;
                nxt[c * LDT + r] = Bp[(k0n + r) * NF + c];
            }
        }

        const float* ap = aRow + (long)s * KB;
        #pragma unroll
        for (int j = 0; j < KB / 4; ++j) {
            // A streamed exactly once -> non-temporal, keep L2 for the B panels.
            v2f a = __builtin_nontemporal_load((const v2f*)(ap + j * 4));
            const int kk = j * 4 + lh * 2;
            #pragma unroll
            for (int nt = 0; nt < 8; ++nt) {
                v2f b = *(const v2f*)(&cur[(nt * 16 + ll) * LDT + kk]);
                acc[nt] = __builtin_amdgcn_wmma_f32_16x16x4_f32(
                    false, a, false, b, (short)0, acc[nt], false, false);
            }
        }
    }

    // Epilogue: fuse row scaling by d_inv and bias add.
    #pragma unroll
    for (int nt = 0; nt < 8; ++nt) {
        const int col = nt * 16 + ll;
        #pragma unroll
        for (int v = 0; v < 8; ++v) {
            const long gRow = gRow0 + v + lh * 8;   // C/D layout: VGPR v, lane half
            float val = acc[nt][v] * dinv[gRow];
            if (bias) val += bias[col];
            out[gRow * NF + col] = val;
        }
    }
}

// ---------------------------------------------------------------------------
// Launch: inputs in setup_inputs order: x, adj, weight, bias (all fp32).
// ---------------------------------------------------------------------------
extern "C" void kernel_launch(void* const* d_in, const int* in_sizes, int n_in,
                              void* d_out, int out_size, void* d_ws, size_t ws_size,
                              hipStream_t stream) {
    const float* x      = (const float*)d_in[0];   // [4,4096,128]
    const float* adj    = (const float*)d_in[1];   // [4,4096,4096]
    const float* weight = (const float*)d_in[2];   // [128,128]
    const float* bias   = (const float*)d_in[3];   // [128]
    float*       out    = (float*)d_out;           // [4,4096,128]

    float* dinv = (float*)d_ws;                    // 4*4096 floats (64 KB)
    float* y    = dinv + 4 * NROWS;                // 4*4096*128 floats (8 MB)

    const int B = 4;

    // 1) d_inv = rsqrt(row-sums of adj)
    gcn_degree<<<(B * NROWS) / 8, 256, 0, stream>>>(adj, dinv);

    // 2) y = diag(d_inv) * (x @ W)      (K=128, W shared across batches)
    gcn_gemm<<<B * (NROWS / 128), 256, 0, stream>>>(
        x, NF, (long)NROWS * NF, weight, 0, NF, dinv, nullptr, y);

    // 3) out = diag(d_inv) * (adj @ y) + bias    (K=4096, the 17-GFLOP GEMM)
    gcn_gemm<<<B * (NROWS / 128), 256, 0, stream>>>(
        adj, NROWS, (long)NROWS * NROWS, y, (long)NROWS * NF, NROWS,
        dinv, bias, out);
}